// My_loss_13494787244319
// MI455X (gfx1250) — compile-verified
//
#include <hip/hip_runtime.h>
#include <math.h>

// Problem constants from the reference
#define B_   32
#define T_   32
#define PER_ 32
#define D_   256
#define N_   (T_ * PER_)

typedef float v2f __attribute__((ext_vector_type(2)));
typedef float v4f __attribute__((ext_vector_type(4)));
typedef float v8f __attribute__((ext_vector_type(8)));

// One wave (32 lanes) per (scene b, frame t).
// Computes the 32x16 Gram tile of the frame's node features via
// V_WMMA_F32_16X16X4_F32 (full f32, matching reference numerics), then the
// masked logsumexp / target-match CE + MSE, then per-frame segment sums via
// wave shuffles. Writes (frame_loss_if_valid, valid_flag) per (b,t) to ws.
__global__ __launch_bounds__(32) void frame_loss_kernel(
    const float* __restrict__ x,      // [B,N,D]
    const float* __restrict__ conf,   // [B,N]
    const int*   __restrict__ ids,    // [B,N]
    const float* __restrict__ scores, // [B,N]
    const int*   __restrict__ tb,     // [B,N] (layout: tb[b, t*PER+j] == t)
    const int*   __restrict__ mib,    // [B,N]
    float* __restrict__ ws)           // [B*T, 2]
{
    __shared__ float sim_lds[32 * 17];   // 32 rows x 16 cols, stride 17 (pad)
    __shared__ float invn[32];

    const int bt   = blockIdx.x;
    const int b    = bt / T_;
    const int t    = bt % T_;
    const int lane = threadIdx.x;        // 0..31, wave32
    const int c    = lane & 15;          // column / row-within-tile
    const int h    = lane >> 4;          // K-half selector

    const long nodeBase = (long)b * N_ + (long)t * PER_;
    const float* __restrict__ xb = x + nodeBase * D_;
    (void)tb; // frame membership is positional in this layout

    // ---- per-node inverse norms (lane = node) ----
    float ss = 0.0f;
    const v4f* __restrict__ xrow = (const v4f*)(xb + lane * D_);
    #pragma unroll 8
    for (int j = 0; j < D_ / 4; ++j) {
        v4f v = xrow[j];
        ss += v.x * v.x + v.y * v.y + v.z * v.z + v.w * v.w;
    }
    invn[lane] = rsqrtf(ss);

    // ---- 32x16x256 Gram tile via f32 WMMA ----
    // A layout (16x4 f32): lane<16 -> row=c, K={k0,k0+1}; lane>=16 -> row=c, K={k0+2,k0+3}
    // B fragment for cols 0..15 is identical to the A fragment of rows 0..15.
    v8f acc0 = {0.f,0.f,0.f,0.f,0.f,0.f,0.f,0.f}; // rows 0..15
    v8f acc1 = acc0;                               // rows 16..31
    const float* __restrict__ pa0 = xb + (c)      * D_ + 2 * h;
    const float* __restrict__ pa1 = xb + (c + 16) * D_ + 2 * h;
    #pragma unroll 4
    for (int k0 = 0; k0 < D_; k0 += 4) {
        v2f a0 = *(const v2f*)(pa0 + k0);
        v2f a1 = *(const v2f*)(pa1 + k0);
        acc0 = __builtin_amdgcn_wmma_f32_16x16x4_f32(
            false, a0, false, a0, (short)0, acc0, false, false);
        acc1 = __builtin_amdgcn_wmma_f32_16x16x4_f32(
            false, a1, false, a0, (short)0, acc1, false, false);
    }

    // ---- spill C tiles to LDS ----
    // C layout: lane holds column c; VGPR r -> row (h*8 + r) within the tile.
    #pragma unroll
    for (int r = 0; r < 8; ++r) {
        sim_lds[(h * 8 + r)      * 17 + c] = acc0[r];
        sim_lds[(16 + h * 8 + r) * 17 + c] = acc1[r];
    }
    __syncthreads();

    // ---- per-row CE / MSE (lane = row) ----
    const int  row  = lane;
    const long nIdx = nodeBase + row;
    const int  idr  = ids[nIdx];
    const int  mir  = mib[nIdx];
    const bool validr = (idr >= 0);
    const bool detr   = validr && (mir != 0);
    const bool gtr    = validr && (mir == 0);
    const float invr  = invn[row];

    float sj[16];
    int   cidA[16];
    unsigned cvmask = 0u;
    float m = -3.0e38f;
    #pragma unroll
    for (int j = 0; j < 16; ++j) {
        const int cid = ids[nodeBase + j];
        const int cmi = mib[nodeBase + j];
        const bool cv = (cid >= 0) && (cmi == 0);   // valid GT column
        const float s = 10.0f * invr * invn[j] * sim_lds[row * 17 + j];
        sj[j]   = s;
        cidA[j] = cid;
        if (cv) { cvmask |= (1u << j); m = fmaxf(m, s); }
    }
    float se = 0.0f, tgt = 0.0f;
    #pragma unroll
    for (int j = 0; j < 16; ++j) {
        if (cvmask & (1u << j)) {
            se += __expf(sj[j] - m);
            if (cidA[j] == idr) tgt += sj[j];
        }
    }
    const float lse = (cvmask != 0u) ? (m + __logf(se)) : -1.0e30f;

    float ce   = detr ? (lse - tgt) : 0.0f;
    float dcf  = conf[nIdx] - scores[nIdx];
    float mse  = detr ? (dcf * dcf) : 0.0f;
    float detf = detr   ? 1.0f : 0.0f;
    float gtf  = gtr    ? 1.0f : 0.0f;
    float vf   = validr ? 1.0f : 0.0f;

    // ---- wave32 segment reductions ----
    #pragma unroll
    for (int off = 16; off > 0; off >>= 1) {
        detf += __shfl_xor(detf, off, 32);
        gtf  += __shfl_xor(gtf,  off, 32);
        vf   += __shfl_xor(vf,   off, 32);
        ce   += __shfl_xor(ce,   off, 32);
        mse  += __shfl_xor(mse,  off, 32);
    }
    if (lane == 0) {
        const bool present = (vf > 0.0f);
        const bool vframe  = present && (detf > 0.0f) && (gtf > 0.0f);
        const float denom  = fmaxf(detf, 1.0f);
        const float fl     = (ce + mse) / denom;
        ws[bt * 2 + 0] = vframe ? fl : 0.0f;
        ws[bt * 2 + 1] = vframe ? 1.0f : 0.0f;
    }
}

// Final reduction: lane = scene; mean over scenes of (sum valid frame losses /
// max(#valid frames, 1)).
__global__ __launch_bounds__(32) void scene_reduce_kernel(
    const float* __restrict__ ws, float* __restrict__ out)
{
    const int b = threadIdx.x;   // 0..31 == B_
    float ls = 0.0f, cnt = 0.0f;
    #pragma unroll 4
    for (int t = 0; t < T_; ++t) {
        ls  += ws[(b * T_ + t) * 2 + 0];
        cnt += ws[(b * T_ + t) * 2 + 1];
    }
    float scene = ls / fmaxf(cnt, 1.0f);
    #pragma unroll
    for (int off = 16; off > 0; off >>= 1)
        scene += __shfl_xor(scene, off, 32);
    if (b == 0) out[0] = scene / (float)B_;
}

extern "C" void kernel_launch(void* const* d_in, const int* in_sizes, int n_in,
                              void* d_out, int out_size, void* d_ws, size_t ws_size,
                              hipStream_t stream) {
    (void)in_sizes; (void)n_in; (void)out_size; (void)ws_size;
    const float* x      = (const float*)d_in[0];
    const float* conf   = (const float*)d_in[1];
    const int*   ids    = (const int*)  d_in[2];
    const float* scores = (const float*)d_in[3];
    const int*   tb     = (const int*)  d_in[4];
    const int*   mib    = (const int*)  d_in[5];
    float* out = (float*)d_out;
    float* ws  = (float*)d_ws;   // B*T*2 floats = 8 KB

    frame_loss_kernel<<<B_ * T_, 32, 0, stream>>>(x, conf, ids, scores, tb, mib, ws);
    scene_reduce_kernel<<<1, 32, 0, stream>>>(ws, out);
}